// Model_1735166788401
// MI455X (gfx1250) — compile-verified
//
#include <hip/hip_runtime.h>
#include <math.h>

// ---- shapes (fixed by the reference harness) ----
#define S      4096     // sequence length
#define H      16       // heads
#define P      64       // state dim p
#define NN     64       // state dim n
#define L      64       // BLOCK_LEN (also the staging tile depth)
#define CHUNKS (S / L)  // 64
#define SPLITK 8        // split-K factor over t
#define TSPAN  (S / SPLITK)   // 512 timesteps per block
#define NCH    (TSPAN / L)    // 8 staged chunks per block
#define LDSP   80       // padded LDS row stride (floats): conflict-free frag loads

typedef __attribute__((ext_vector_type(2))) float v2f;
typedef __attribute__((ext_vector_type(4))) float v4f;
typedef __attribute__((ext_vector_type(8))) float v8f;
typedef __attribute__((ext_vector_type(4))) int   v4i;

// Async global->LDS staging path (CDNA5): guarded so the kernel still
// compiles (synchronous fallback) if this toolchain lacks the builtins.
#if defined(__has_builtin)
#if __has_builtin(__builtin_amdgcn_global_load_async_to_lds_b128) && \
    __has_builtin(__builtin_amdgcn_s_wait_asynccnt)
#define USE_ASYNC_LDS 1
#endif
#endif
#ifndef USE_ASYNC_LDS
#define USE_ASYNC_LDS 0
#endif

// builtin signature: (v4i AS1* gsrc, v4i AS3* ldst, imm offset, imm cpol)
#define AS_GLB(p) ((__attribute__((address_space(1))) v4i*)(void*)(const void*)(p))
#define AS_LDS(p) ((__attribute__((address_space(3))) v4i*)(p))

// --------------------------------------------------------------------------
// Kernel 0: zero the output (we accumulate split-K partials with f32 atomics)
// --------------------------------------------------------------------------
__global__ void zero_kernel(float* __restrict__ out, int n) {
    int i = blockIdx.x * blockDim.x + threadIdx.x;
    if (i < n) out[i] = 0.0f;
}

// --------------------------------------------------------------------------
// Kernel 1: per-(b,h) suffix-exp weights
//   w[bh][t] = exp( sum_{t' > t} A[b, t', h] )
// --------------------------------------------------------------------------
__global__ void __launch_bounds__(CHUNKS)
weights_kernel(const float* __restrict__ A, float* __restrict__ w) {
    const int bh = blockIdx.x;              // 0..63
    const int bb = bh >> 4;
    const int hh = bh & 15;
    const float* Ap = A + (size_t)bb * S * H + hh;   // element t at Ap[t*H]
    float* wp = w + (size_t)bh * S;

    const int j = threadIdx.x;              // chunk index 0..63
    const int tbase = j * L;

    __shared__ float tot[CHUNKS];

    float tsum = 0.0f;
    for (int i = 0; i < L; ++i) tsum += Ap[(size_t)(tbase + i) * H];
    tot[j] = tsum;
    __syncthreads();

    float suf = 0.0f;
    for (int k = j + 1; k < CHUNKS; ++k) suf += tot[k];

    float run = 0.0f;
    for (int i = L - 1; i >= 0; --i) {
        float a = Ap[(size_t)(tbase + i) * H];
        wp[tbase + i] = expf(run + suf);
        run += a;
    }
}

// --------------------------------------------------------------------------
// Kernel 2: per-(b,h) GEMM  out[p,n] += sum_t (w_t * X[t,p]) * B[t,n]
// grid (64 bh, 8 splitK); 128 threads = 4 wave32's.
// Double-buffered LDS tiles staged with GLOBAL_LOAD_ASYNC_TO_LDS_B128
// (ASYNCcnt pipelined against the WMMA loop); V_WMMA_F32_16X16X4_F32.
// --------------------------------------------------------------------------
__global__ void __launch_bounds__(128)
ssd_gemm_kernel(const float* __restrict__ X, const float* __restrict__ B,
                const float* __restrict__ w, float* __restrict__ out) {
    const int bh    = blockIdx.x;
    const int split = blockIdx.y;
    const int bb = bh >> 4;
    const int hh = bh & 15;

    // element (t,p): Xp + t*(H*P) + p     (t-stride = 1024 floats)
    const float* Xp = X + ((size_t)bb * S * H + hh) * P;
    const float* Bp = B + ((size_t)bb * S * H + hh) * NN;
    const float* wp = w + (size_t)bh * S;
    float* outp = out + (size_t)bh * P * NN;

    __shared__ float Xs[2][L * LDSP];   // raw X tile [t][p], padded
    __shared__ float Bs[2][L * LDSP];   // B tile [t][n], padded
    __shared__ float Ws[2][L];          // per-t weights for the tile

    const int tid  = threadIdx.x;
    const int wave = tid >> 5;       // 0..3 -> M tile
    const int lane = tid & 31;
    const int lm   = lane & 15;      // M/N index within tile
    const int hi   = lane >> 4;      // 0: K rows {kb,kb+1}, 1: {kb+2,kb+3}

    const int r = tid >> 4;          // staging: t-row 0..7 handled by this thread (x8)
    const int q = (tid & 15) << 2;   // staging: col 0,4,...,60

    v8f acc0 = {}, acc1 = {}, acc2 = {}, acc3 = {};

    const int t0 = split * TSPAN;

    // ---- staging: one 64x64 tile of X and of B (+64 w's) into buffer `bf` ----
    auto issue_stage = [&](int tc, int bf) {
        #pragma unroll
        for (int u = 0; u < 8; ++u) {
            const int rr = r + 8 * u;                       // t-row 0..63
            const float* gx = Xp + (size_t)(tc + rr) * (H * P) + q;
            const float* gb = Bp + (size_t)(tc + rr) * (H * NN) + q;
            float* lx = &Xs[bf][rr * LDSP + q];
            float* lb = &Bs[bf][rr * LDSP + q];
#if USE_ASYNC_LDS
            __builtin_amdgcn_global_load_async_to_lds_b128(AS_GLB(gx), AS_LDS(lx), 0, 0);
            __builtin_amdgcn_global_load_async_to_lds_b128(AS_GLB(gb), AS_LDS(lb), 0, 0);
#else
            *(v4f*)lx = *(const v4f*)gx;
            *(v4f*)lb = *(const v4f*)gb;
#endif
        }
        if (tid < L) Ws[bf][tid] = wp[tc + tid];            // tiny, regular ds store
    };

    issue_stage(t0, 0);                                      // prologue: chunk 0

    for (int c = 0; c < NCH; ++c) {
        const int cb = c & 1;

        // prefetch next chunk into the other buffer, then wait for chunk c
        if (c + 1 < NCH) {
            issue_stage(t0 + (c + 1) * L, (c + 1) & 1);
#if USE_ASYNC_LDS
            __builtin_amdgcn_s_wait_asynccnt(16);            // chunk c done; c+1 in flight
#endif
        } else {
#if USE_ASYNC_LDS
            __builtin_amdgcn_s_wait_asynccnt(0);
#endif
        }
        __syncthreads();                                     // chunk c visible to all waves

        // ---- 16 K-steps of V_WMMA_F32_16X16X4_F32 on buffer cb ----
        // A 16x4 layout: lane(M=lm) holds K = 2*hi + {0,1}; B mirrored with N=lm.
        // Weight w_t applied to the A fragment (broadcast LDS reads).
        #pragma unroll
        for (int kb = 0; kb < L; kb += 4) {
            const int rowA = kb + 2 * hi;
            const float w0 = Ws[cb][rowA];
            const float w1 = Ws[cb][rowA + 1];
            v2f a;
            a.x = Xs[cb][rowA * LDSP + 16 * wave + lm] * w0;
            a.y = Xs[cb][(rowA + 1) * LDSP + 16 * wave + lm] * w1;
            v2f b0, b1, b2, b3;
            b0.x = Bs[cb][rowA * LDSP +  0 + lm];  b0.y = Bs[cb][(rowA + 1) * LDSP +  0 + lm];
            b1.x = Bs[cb][rowA * LDSP + 16 + lm];  b1.y = Bs[cb][(rowA + 1) * LDSP + 16 + lm];
            b2.x = Bs[cb][rowA * LDSP + 32 + lm];  b2.y = Bs[cb][(rowA + 1) * LDSP + 32 + lm];
            b3.x = Bs[cb][rowA * LDSP + 48 + lm];  b3.y = Bs[cb][(rowA + 1) * LDSP + 48 + lm];
            acc0 = __builtin_amdgcn_wmma_f32_16x16x4_f32(false, a, false, b0, (short)0, acc0, false, false);
            acc1 = __builtin_amdgcn_wmma_f32_16x16x4_f32(false, a, false, b1, (short)0, acc1, false, false);
            acc2 = __builtin_amdgcn_wmma_f32_16x16x4_f32(false, a, false, b2, (short)0, acc2, false, false);
            acc3 = __builtin_amdgcn_wmma_f32_16x16x4_f32(false, a, false, b3, (short)0, acc3, false, false);
        }
        __syncthreads();   // all waves done reading buffer cb before it is overwritten
    }

    // ---- combine split-K partials: f32 global atomics (16 KB per block) ----
    // C/D layout: VGPR v -> M = v + 8*hi (lanes 16-31 are +8), N = lm.
    #pragma unroll
    for (int v = 0; v < 8; ++v) {
        const int p = 16 * wave + v + 8 * hi;
        atomicAdd(&outp[(size_t)p * NN +  0 + lm], acc0[v]);
        atomicAdd(&outp[(size_t)p * NN + 16 + lm], acc1[v]);
        atomicAdd(&outp[(size_t)p * NN + 32 + lm], acc2[v]);
        atomicAdd(&outp[(size_t)p * NN + 48 + lm], acc3[v]);
    }
}

// --------------------------------------------------------------------------
extern "C" void kernel_launch(void* const* d_in, const int* in_sizes, int n_in,
                              void* d_out, int out_size, void* d_ws, size_t ws_size,
                              hipStream_t stream) {
    const float* X = (const float*)d_in[0];   // (b, s, h, p) f32
    const float* A = (const float*)d_in[1];   // (b, s, h)    f32
    const float* B = (const float*)d_in[2];   // (b, s, h, n) f32
    // d_in[3] = C : unused by the reference computation
    float* out = (float*)d_out;               // (b, h, p, n) f32
    float* wws = (float*)d_ws;                // 64 * 4096 floats = 1 MiB

    zero_kernel<<<(out_size + 255) / 256, 256, 0, stream>>>(out, out_size);
    weights_kernel<<<64, CHUNKS, 0, stream>>>(A, wws);
    ssd_gemm_kernel<<<dim3(64, SPLITK), 128, 0, stream>>>(X, B, wws, out);
}